// Attention_61959198212370
// MI455X (gfx1250) — compile-verified
//
#include <hip/hip_runtime.h>
#include <hip/hip_bf16.h>

typedef float v2f __attribute__((ext_vector_type(2)));
typedef float v8f __attribute__((ext_vector_type(8)));
typedef int   v4i __attribute__((vector_size(16)));   // matches builtin param type

#define N_B    16
#define NQ     64
#define NV     128
#define NH     4
#define NE     128
#define NHE    (NH * NE)      // 512
#define QT     16             // queries per block
#define THREADS 512           // 16 wave32s

// Padded LDS strides (floats). Stride 132: consecutive rows land 132 apart
// -> 132 % 64 = 4 banks, so row-parallel access patterns don't alias.
// 132 floats = 528 B = multiple of 16 B, so rows stay b128-aligned.
#define ESTR   132

__device__ __forceinline__ float my_tanh(float x) {
#if defined(__AMDGCN__) && __has_builtin(__builtin_amdgcn_tanhf)
    return __builtin_amdgcn_tanhf(x);   // gfx1250 v_tanh_f32
#else
    return tanhf(x);
#endif
}

__device__ __forceinline__ float leaky(float x) {
    return x > 0.0f ? x : 0.01f * x;
}

// ---- gfx1250 async global->LDS copy (ASYNCcnt-tracked DMA), with safe fallback ----
__device__ __forceinline__ void async_g2l_b128(const float* g, float* l) {
#if defined(__AMDGCN__) && __has_builtin(__builtin_amdgcn_global_load_async_to_lds_b128)
    // Builtin signature (from compiler diagnostic): (global v4i*, local v4i*, imm, imm).
    // Generic->AS casts via uintptr_t; low 32 bits of a generic LDS pointer are
    // the wave-relative LDS byte offset (aperture rule).
    __builtin_amdgcn_global_load_async_to_lds_b128(
        (__attribute__((address_space(1))) v4i*)(uintptr_t)g,
        (__attribute__((address_space(3))) v4i*)(uint32_t)(uintptr_t)l,
        0, 0);
#else
    *(float4*)l = *(const float4*)g;    // synchronous fallback
#endif
}

__device__ __forceinline__ void wait_async_done() {
#if defined(__AMDGCN__) && __has_builtin(__builtin_amdgcn_s_wait_asynccnt)
    __builtin_amdgcn_s_wait_asynccnt(0);
#endif
}

__global__ __launch_bounds__(THREADS)
void attn_fused_wmma(const float* __restrict__ query,      // (16,64,128)
                     const float* __restrict__ context,    // (16,128,128)
                     const float* __restrict__ memory,     // (16,128,128)
                     const float* __restrict__ mask,       // (16,128)
                     const float* __restrict__ W_create,   // (512,128) row-major
                     const float* __restrict__ b_create,   // (512)
                     const float* __restrict__ w_logit,    // (128)
                     const float* __restrict__ b_logit,    // (1)
                     const float* __restrict__ W_reduce,   // (128,512) row-major
                     const float* __restrict__ b_reduce,   // (128)
                     const float* __restrict__ temperature,// (1)
                     float* __restrict__ out)              // (16,64,128)
{
    __shared__ float ctxL [NV * ESTR];         // 66 KB  context[v][e], padded
    __shared__ float memL [NV * ESTR];         // 66 KB  memory[v][e], padded
    __shared__ float qprojL[QT * NH * ESTR];   // 33.8 KB [q][h][e]
    __shared__ float probsL[QT * NH * ESTR];   // 33.8 KB logits -> probs in place
    __shared__ float headsL[QT * NH * ESTR];   // 33.8 KB [q][h][e]
    __shared__ float wlogL[NE];
    __shared__ float maskL[NV];

    const int b    = blockIdx.x >> 2;          // batch
    const int q0   = (blockIdx.x & 3) * QT;    // query tile base
    const int t    = threadIdx.x;
    const int wave = t >> 5;
    const int lane = t & 31;
    const int lo   = lane & 15;                // M / N index within fragment
    const int hi   = lane >> 4;                // half-wave select

    // ---- kick off async DMA of context & memory into LDS (overlaps phase 1) ----
    {
        const float* cg = context + (size_t)b * NV * NE;
        const float* mg = memory  + (size_t)b * NV * NE;
        for (int i = t; i < (NV * NE) / 4; i += THREADS) {   // 4096 x 16B chunks each
            const int v = i >> 5;            // row
            const int c = (i & 31) * 4;      // float offset within row
            async_g2l_b128(cg + v * NE + c, ctxL + v * ESTR + c);
            async_g2l_b128(mg + v * NE + c, memL + v * ESTR + c);
        }
    }

    // ---- stage small vectors; prefetch W_reduce (consumed last) into L2 ----
    if (t < NE)                 wlogL[t]      = w_logit[t];
    else if (t < NE + NV)       maskL[t - NE] = mask[b * NV + (t - NE)];
    for (int i = t; i < (NHE * NE) / 64; i += THREADS)       // 256KB, 256B strides
        __builtin_prefetch(W_reduce + i * 64, 0, 1);         // global_prefetch_b8

    // =========== Phase 1: qproj(16x512) = queryTile(16x128) @ W_create^T + b ===========
    // WMMA f32 16x16x4: A[m][k] -> lane(lo)=m, a.{x,y} = k0+2*hi {+0,+1}
    //                   B[k][n] -> lane(lo)=n, b.{x,y} = rows k0+2*hi {+0,+1}
    //                   D: vgpr r -> m = r + 8*hi, n = lo
    {
        const float* qbase = query + (size_t)(b * NQ + q0) * NE;
        for (int tile = wave; tile < NHE / 16; tile += THREADS / 32) {
            const int n = tile * 16 + lo;        // output channel in [0,512)
            v8f acc = {};
            for (int k0 = 0; k0 < NE; k0 += 4) {
                const int ka = k0 + 2 * hi;
                v2f a, bb;
                a.x  = qbase[lo * NE + ka];
                a.y  = qbase[lo * NE + ka + 1];
                bb.x = W_create[n * NE + ka];       // B[k][n] = W_create[n][k]
                bb.y = W_create[n * NE + ka + 1];
                acc = __builtin_amdgcn_wmma_f32_16x16x4_f32(
                        false, a, false, bb, (short)0, acc, false, false);
            }
            const float bc = b_create[n];
            const int h = n >> 7, e = n & 127;
            #pragma unroll
            for (int r = 0; r < 8; ++r) {
                const int m = r + 8 * hi;
                qprojL[(m * NH + h) * ESTR + e] = acc[r] + bc;
            }
        }
    }
    wait_async_done();     // per-wave: all our async LDS writes landed
    __syncthreads();       // cross-wave visibility of ctxL/memL/qprojL

    // =========== Phase 2: logits[q,v,h] = sum_e tanh(qproj[q,h,e]+ctx[v,e])*wl[e] ======
    // One (v,h) pair per thread; 16 tanh+fma amortize each ctx/wlog LDS load.
    // ctxL v-stride 132 -> 8 distinct v per wave map to 8 distinct banks.
    {
        const int v = t >> 2;
        const int h = t & 3;
        const float* cb = ctxL + v * ESTR;
        const float* qp = qprojL + h * ESTR;
        float acc[QT];
        #pragma unroll
        for (int q = 0; q < QT; ++q) acc[q] = 0.0f;

        for (int e = 0; e < NE; ++e) {
            const float c  = cb[e];
            const float wl = wlogL[e];
            #pragma unroll
            for (int q = 0; q < QT; ++q)
                acc[q] += my_tanh(qp[q * NH * ESTR + e] + c) * wl;
        }

        const float m    = maskL[v];
        const float sc   = (1.0f / temperature[0]) * m;
        const float neg  = -99999.0f * (1.0f - m);
        const float blog = b_logit[0];
        #pragma unroll
        for (int q = 0; q < QT; ++q)
            probsL[(q * NH + h) * ESTR + v] = (acc[q] + blog) * sc + neg;
    }
    __syncthreads();

    // =========== Softmax over v per (q,h) row; fold mask into probs =====================
    if (t < QT * NH) {
        float* row = probsL + t * ESTR;
        float mx = row[0];
        for (int v = 1; v < NV; ++v) mx = fmaxf(mx, row[v]);
        float s = 0.0f;
        for (int v = 0; v < NV; ++v) { float e = __expf(row[v] - mx); row[v] = e; s += e; }
        const float inv = 1.0f / s;
        for (int v = 0; v < NV; ++v) row[v] = row[v] * inv * maskL[v];
    }
    __syncthreads();

    // =========== Phase 3: heads[q,h,e] = sum_v probs[q,h,v] * memL[v,e]; leaky_relu =====
    {
        for (int tile = wave; tile < NHE / 16; tile += THREADS / 32) {
            const int h  = tile >> 3;
            const int n  = (tile & 7) * 16 + lo;     // e index
            const float* pb = probsL + h * ESTR;      // A[m=q][k=v], row stride NH*ESTR
            v8f acc = {};
            for (int k0 = 0; k0 < NV; k0 += 4) {
                const int ka = k0 + 2 * hi;
                v2f a, bb;
                a.x  = pb[lo * NH * ESTR + ka];
                a.y  = pb[lo * NH * ESTR + ka + 1];
                bb.x = memL[ka * ESTR + n];           // B[k][n] = memory[v=k][e=n]
                bb.y = memL[(ka + 1) * ESTR + n];
                acc = __builtin_amdgcn_wmma_f32_16x16x4_f32(
                        false, a, false, bb, (short)0, acc, false, false);
            }
            #pragma unroll
            for (int r = 0; r < 8; ++r) {
                const int m = r + 8 * hi;
                headsL[(m * NH + h) * ESTR + n] = leaky(acc[r]);
            }
        }
    }
    __syncthreads();

    // =========== Phase 4: out(16x128) = heads(16x512) @ W_reduce^T + b_reduce ===========
    if (wave < 8) {
        const int n = wave * 16 + lo;                // output channel in [0,128)
        v8f acc = {};
        for (int k0 = 0; k0 < NHE; k0 += 4) {
            const int ka = k0 + 2 * hi;              // even, so ka/ka+1 share an h-block
            const int h = ka >> 7, e = ka & 127;
            v2f a, bb;
            a.x  = headsL[(lo * NH + h) * ESTR + e];
            a.y  = headsL[(lo * NH + h) * ESTR + e + 1];
            bb.x = W_reduce[n * NHE + ka];           // B[k][n] = W_reduce[n][k]
            bb.y = W_reduce[n * NHE + ka + 1];
            acc = __builtin_amdgcn_wmma_f32_16x16x4_f32(
                    false, a, false, bb, (short)0, acc, false, false);
        }
        const float br = b_reduce[n];
        float* ob = out + (size_t)(b * NQ + q0) * NE;
        #pragma unroll
        for (int r = 0; r < 8; ++r) {
            const int m = r + 8 * hi;
            ob[m * NE + n] = acc[r] + br;
        }
    }
}

extern "C" void kernel_launch(void* const* d_in, const int* in_sizes, int n_in,
                              void* d_out, int out_size, void* d_ws, size_t ws_size,
                              hipStream_t stream) {
    const float* query       = (const float*)d_in[0];
    const float* context     = (const float*)d_in[1];
    const float* memory      = (const float*)d_in[2];
    const float* mask        = (const float*)d_in[3];
    const float* W_create    = (const float*)d_in[4];
    const float* b_create    = (const float*)d_in[5];
    const float* w_logit     = (const float*)d_in[6];
    const float* b_logit     = (const float*)d_in[7];
    const float* W_reduce    = (const float*)d_in[8];
    const float* b_reduce    = (const float*)d_in[9];
    const float* temperature = (const float*)d_in[10];
    float* out = (float*)d_out;

    dim3 grid(N_B * (NQ / QT));   // 64 blocks: (batch, 16-query tile)
    dim3 block(THREADS);          // 16 wave32s
    attn_fused_wmma<<<grid, block, 0, stream>>>(
        query, context, memory, mask, W_create, b_create,
        w_logit, b_logit, W_reduce, b_reduce, temperature, out);
}